// RankingLoss_34772055229082
// MI455X (gfx1250) — compile-verified
//
#include <hip/hip_runtime.h>
#include <math.h>

#define BLOCK 256
#define MARGIN_C 0.2f
#define MARGIN_W 0.3f

typedef __attribute__((ext_vector_type(2))) float v2f;
typedef __attribute__((ext_vector_type(4))) float v4f;
typedef __attribute__((ext_vector_type(8))) float v8f;

// Sum v across all 32 lanes of a wave using the CDNA5 matrix pipe.
// WMMA1: A[m,k] with A.x=A.y=v  ->  D[m,n] = 2*(v_m + v_{m+16})
// Per-lane w = sum of 8 D VGPRs is uniform per half-wave (2*S_lo / 2*S_hi).
// WMMA2 on w: D2[m,n] = 2*(w_m + w_{m+16}) = 4 * total, uniform on all lanes.
// Requires EXEC == all ones (call from converged code only).
__device__ __forceinline__ float wave_sum_wmma(float v) {
  v2f one; one[0] = 1.0f; one[1] = 1.0f;
  v8f c = {0.f, 0.f, 0.f, 0.f, 0.f, 0.f, 0.f, 0.f};
  v2f a; a[0] = v; a[1] = v;
  v8f d = __builtin_amdgcn_wmma_f32_16x16x4_f32(false, a, false, one,
                                                (short)0, c, false, false);
  float w = ((d[0] + d[1]) + (d[2] + d[3])) + ((d[4] + d[5]) + (d[6] + d[7]));
  v2f a2; a2[0] = w; a2[1] = w;
  v8f d2 = __builtin_amdgcn_wmma_f32_16x16x4_f32(false, a2, false, one,
                                                 (short)0, c, false, false);
  return d2[0] * 0.25f;
}

// Deterministic block-wide sum of (s, g). Valid results on tid==0 only.
// lds must hold >= 16 floats.
__device__ __forceinline__ void block_sum2(float s, float g, float* S, float* G,
                                           float* lds, int tid) {
  float ws = wave_sum_wmma(s);
  float wg = wave_sum_wmma(g);
  const int wid = tid >> 5;
  if ((tid & 31) == 0) {
    lds[wid] = ws;
    lds[8 + wid] = wg;
  }
  __syncthreads();
  if (tid == 0) {
    float a = 0.f, b = 0.f;
    #pragma unroll
    for (int i = 0; i < BLOCK / 32; ++i) { a += lds[i]; b += lds[8 + i]; }
    *S = a;
    *G = b;
  }
}

// One block per row. Single streamed pass:
//   s = sum_j exp(x_j - correct)          -> ce_row = log(s)
//   g = sum_j relu(x_j - correct + 0.2)   -> row margin sum = g - 0.2 (label term)
__global__ __launch_bounds__(BLOCK) void rank_row_kernel(
    const float* __restrict__ scores, const int* __restrict__ labels,
    float* __restrict__ row_ce, float* __restrict__ row_g, int N) {
  __shared__ float s_red[16];
  __shared__ float s_correct;
  const int tid = threadIdx.x;
  const int b = blockIdx.x;
  const float* __restrict__ row = scores + (size_t)b * (size_t)N;

  if (tid == 0) s_correct = row[labels[b]];
  __syncthreads();
  const float correct = s_correct;

  float s0 = 0.f, s1 = 0.f, g0 = 0.f, g1 = 0.f;

  // Rows are only 4B-aligned (N odd): peel to 16B alignment for b128 loads.
  int head = (int)((16u - (unsigned)((uintptr_t)row & 15u)) & 15u) >> 2;
  if (head > N) head = N;
  if (tid < head) {
    float x = row[tid] - correct;
    s0 += __expf(x);
    g0 += fmaxf(x + MARGIN_C, 0.f);
  }

  const int nv = (N - head) >> 2;
  const v4f* __restrict__ vrow = (const v4f*)(row + head);
  for (int i = tid; i < nv; i += BLOCK) {
    v4f v = __builtin_nontemporal_load(vrow + i);  // global_load_b128 th:NT
    float x0 = v[0] - correct, x1 = v[1] - correct;
    float x2 = v[2] - correct, x3 = v[3] - correct;
    s0 += __expf(x0) + __expf(x2);
    s1 += __expf(x1) + __expf(x3);
    g0 += fmaxf(x0 + MARGIN_C, 0.f) + fmaxf(x2 + MARGIN_C, 0.f);
    g1 += fmaxf(x1 + MARGIN_C, 0.f) + fmaxf(x3 + MARGIN_C, 0.f);
  }

  for (int i = head + nv * 4 + tid; i < N; i += BLOCK) {
    float x = row[i] - correct;
    s1 += __expf(x);
    g1 += fmaxf(x + MARGIN_C, 0.f);
  }

  float S, G;
  block_sum2(s0 + s1, g0 + g1, &S, &G, s_red, tid);
  if (tid == 0) {
    row_ce[b] = logf(S);           // logsumexp(row) - correct
    row_g[b] = G - MARGIN_C;       // drop the label's exact 0.2 contribution
  }
}

__global__ __launch_bounds__(BLOCK) void rank_final_kernel(
    const float* __restrict__ row_ce, const float* __restrict__ row_g,
    float* __restrict__ out, int B, int N) {
  __shared__ float s_red[16];
  const int tid = threadIdx.x;
  float cs = 0.f, gs = 0.f;
  for (int i = tid; i < B; i += BLOCK) {
    cs += row_ce[i];
    gs += row_g[i];
  }
  float CS, GS;
  block_sum2(cs, gs, &CS, &GS, s_red, tid);
  if (tid == 0) {
    double ce = (double)CS / (double)B;
    double margin = (double)GS / ((double)B * (double)N);
    out[0] = (float)(ce + (double)MARGIN_W * margin);
    out[1] = (float)ce;
    out[2] = (float)margin;
  }
}

extern "C" void kernel_launch(void* const* d_in, const int* in_sizes, int n_in,
                              void* d_out, int out_size, void* d_ws, size_t ws_size,
                              hipStream_t stream) {
  const float* scores = (const float*)d_in[0];
  const int* labels = (const int*)d_in[1];
  const int B = in_sizes[1];
  const int N = in_sizes[0] / B;

  float* row_ce = (float*)d_ws;
  float* row_g = row_ce + B;
  float* out = (float*)d_out;

  rank_row_kernel<<<B, BLOCK, 0, stream>>>(scores, labels, row_ce, row_g, N);
  rank_final_kernel<<<1, BLOCK, 0, stream>>>(row_ce, row_g, out, B, N);
}